// PointConvD_5291399708681
// MI455X (gfx1250) — compile-verified
//
#include <hip/hip_runtime.h>

typedef __attribute__((ext_vector_type(2))) float v2f;
typedef __attribute__((ext_vector_type(8))) float v8f;

#define B_     8
#define N_     8192
#define S_     2048
#define D_     64
#define CIN_   67
#define CPAD_  68
#define WN_    16
#define F_     1072      // CIN_*WN_
#define OUT_   128
#define LEAKY_ 0.1f
#define TILE_  1024
#define FMAXV  3.402823466e+38f

// ---------------------------------------------------------------------------
// Kernel 1: furthest point sampling. One workgroup (1024 thr = 32 waves) per
// batch. Each thread owns 8 points in registers; per-step argmax via shuffle
// reduction + cross-wave LDS reduction. Writes new_xyz in [B,3,S] layout
// (which is also the first chunk of d_out).
// ---------------------------------------------------------------------------
__global__ __launch_bounds__(1024) void fps_kernel(const float* __restrict__ xyz,
                                                   float* __restrict__ newXyz) {
  const int b    = blockIdx.x;
  const int tid  = threadIdx.x;
  const int lane = tid & 31;
  const int wid  = tid >> 5;
  const float* xb = xyz + (size_t)b * 3 * N_;

  float px[8], py[8], pz[8], dist[8];
#pragma unroll
  for (int j = 0; j < 8; ++j) {
    int p   = tid + 1024 * j;
    px[j]   = xb[p];
    py[j]   = xb[N_ + p];
    pz[j]   = xb[2 * N_ + p];
    dist[j] = __builtin_huge_valf();
  }

  __shared__ float s_rv[32];
  __shared__ int   s_ri[32];
  __shared__ int   s_far;

  int far = 0;
  float* outb = newXyz + (size_t)b * 3 * S_;

  for (int it = 0; it < S_; ++it) {
    float cx = xb[far], cy = xb[N_ + far], cz = xb[2 * N_ + far];
    if (tid == 0) { outb[it] = cx; outb[S_ + it] = cy; outb[2 * S_ + it] = cz; }

    float bv = -1.0f; int bidx = 0;
#pragma unroll
    for (int j = 0; j < 8; ++j) {
      float dx = px[j] - cx, dy = py[j] - cy, dz = pz[j] - cz;
      float d  = dx * dx + dy * dy + dz * dz;
      float nd = fminf(dist[j], d);
      dist[j]  = nd;
      if (nd > bv) { bv = nd; bidx = tid + 1024 * j; }   // j ascending => lowest idx first
    }
#pragma unroll
    for (int off = 16; off > 0; off >>= 1) {
      float ov = __shfl_xor(bv, off);
      int   oi = __shfl_xor(bidx, off);
      if (ov > bv || (ov == bv && oi < bidx)) { bv = ov; bidx = oi; }
    }
    if (lane == 0) { s_rv[wid] = bv; s_ri[wid] = bidx; }
    __syncthreads();
    if (wid == 0) {
      float v  = s_rv[lane];
      int   i2 = s_ri[lane];
#pragma unroll
      for (int off = 16; off > 0; off >>= 1) {
        float ov = __shfl_xor(v, off);
        int   oi = __shfl_xor(i2, off);
        if (ov > v || (ov == v && oi < i2)) { v = ov; i2 = oi; }
      }
      if (lane == 0) s_far = i2;
    }
    __syncthreads();
    far = s_far;
  }
}

// ---------------------------------------------------------------------------
// Kernel 2 (fused): KNN top-16 -> WeightNet MLP -> weighted aggregation ->
// final linear (WMMA f32 16x16x4) + bias + LeakyReLU + transposed store.
// One wave per center; block = 8 waves = 8 centers (all in the same batch).
// LDS union region reused across phases:
//   phase A: xyz tile (1024 pts * float4, with |p|^2 in .w)     = 4096 f
//   phase B: per-wave chunk {np[16][68], wgt[16][16]} = 1344 f  = 10752 f
//   phase C: agg[8][1072]                                       = 8576 f
// ---------------------------------------------------------------------------
__global__ __launch_bounds__(256) void pointconv_kernel(
    const float* __restrict__ xyz,  const float* __restrict__ pts,
    const float* __restrict__ w0g,  const float* __restrict__ b0g,
    const float* __restrict__ w1g,  const float* __restrict__ b1g,
    const float* __restrict__ w2g,  const float* __restrict__ b2g,
    const float* __restrict__ lw,   const float* __restrict__ lb,
    const float* __restrict__ newXyz, float* __restrict__ outPts) {
  const int tid        = threadIdx.x;
  const int lane       = tid & 31;
  const int wid        = tid >> 5;
  const int centerBase = blockIdx.x * 8;
  const int center     = centerBase + wid;
  const int b          = center >> 11;          // S_=2048 per batch
  const int s          = center & (S_ - 1);
  const float* xb = xyz + (size_t)b * 3 * N_;
  const float* pb = pts + (size_t)b * D_ * N_;

  __shared__ float s_mlp[256];
  __shared__ float s_union[10752];

  // stage WeightNet parameters (barrier provided by first tile-loop sync)
  if      (tid < 24)  s_mlp[tid] = w0g[tid];
  else if (tid < 32)  s_mlp[tid] = b0g[tid - 24];
  else if (tid < 96)  s_mlp[tid] = w1g[tid - 32];
  else if (tid < 104) s_mlp[tid] = b1g[tid - 96];
  else if (tid < 232) s_mlp[tid] = w2g[tid - 104];
  else if (tid < 248) s_mlp[tid] = b2g[tid - 232];

  const float* nxb = newXyz + (size_t)b * 3 * S_;
  const float cx = nxb[s], cy = nxb[S_ + s], cz = nxb[2 * S_ + s];
  const float sn = cx * cx + cy * cy + cz * cz;

  // --- phase A: scan all N points, per-lane sorted top-16 (dist, idx) ------
  float bd[16]; int bi[16];
#pragma unroll
  for (int t = 0; t < 16; ++t) { bd[t] = FMAXV; bi[t] = 0x7fffffff; }

  float* s_tile = s_union;
  for (int tb = 0; tb < N_; tb += TILE_) {
    __syncthreads();
#pragma unroll
    for (int q = 0; q < TILE_ / 256; ++q) {
      int l = tid + 256 * q;
      int p = tb + l;
      float x = xb[p], y = xb[N_ + p], z = xb[2 * N_ + p];
      s_tile[l * 4 + 0] = x; s_tile[l * 4 + 1] = y; s_tile[l * 4 + 2] = z;
      s_tile[l * 4 + 3] = x * x + y * y + z * z;
    }
    __syncthreads();
    for (int jj = 0; jj < TILE_ / 32; ++jj) {
      int   l  = lane + 32 * jj;
      float x  = s_tile[l * 4 + 0], y = s_tile[l * 4 + 1];
      float z  = s_tile[l * 4 + 2], pn = s_tile[l * 4 + 3];
      float dot = cx * x + cy * y + cz * z;
      float d   = -2.0f * dot + sn + pn;   // matches reference expanded form
      int   p   = tb + l;
      if (d < bd[15] || (d == bd[15] && p < bi[15])) {
        float dd = d; int ii = p;
#pragma unroll
        for (int t = 0; t < 16; ++t) {
          bool  sw = (dd < bd[t]) || (dd == bd[t] && ii < bi[t]);
          float td = sw ? bd[t] : dd;  int ti = sw ? bi[t] : ii;
          bd[t] = sw ? dd : bd[t];     bi[t] = sw ? ii : bi[t];
          dd = td; ii = ti;
        }
      }
    }
  }
  __syncthreads();   // tile region free for reuse after this

  // --- merge 32 sorted lists -> global top-16 via head-compare shuffles ----
  int myIdx = 0;     // neighbor index for k = lane&15
#pragma unroll
  for (int r = 0; r < 16; ++r) {
    float hd = bd[0]; int hi2 = bi[0]; int wl = lane;
#pragma unroll
    for (int off = 16; off > 0; off >>= 1) {
      float od = __shfl_xor(hd, off);
      int   oi = __shfl_xor(hi2, off);
      int   ol = __shfl_xor(wl, off);
      if (od < hd || (od == hd && oi < hi2)) { hd = od; hi2 = oi; wl = ol; }
    }
    if ((lane & 15) == r) myIdx = hi2;
    if (lane == wl) {
#pragma unroll
      for (int t = 0; t < 15; ++t) { bd[t] = bd[t + 1]; bi[t] = bi[t + 1]; }
      bd[15] = FMAXV; bi[15] = 0x7fffffff;
    }
  }

  // --- phase B: gather neighbor features + WeightNet MLP -------------------
  float* s_np  = s_union + wid * 1344;   // [16][68]
  float* s_wgt = s_np + 1088;            // [16][16]
  const int k    = lane & 15;
  const int half = lane >> 4;

  float nx = xb[myIdx], ny = xb[N_ + myIdx], nz = xb[2 * N_ + myIdx];
  float gx = nx - cx, gy = ny - cy, gz = nz - cz;
  if (half == 0) {
    s_np[k * CPAD_ + 0] = gx; s_np[k * CPAD_ + 1] = gy; s_np[k * CPAD_ + 2] = gz;
  }
#pragma unroll
  for (int q = 0; q < 32; ++q) {
    int dfeat = half * 32 + q;
    s_np[k * CPAD_ + 3 + dfeat] = pb[(size_t)dfeat * N_ + myIdx];
  }
  if (half == 0) {
    const float *W0 = s_mlp, *B0 = s_mlp + 24, *W1 = s_mlp + 32,
                *B1 = s_mlp + 96, *W2 = s_mlp + 104, *B2 = s_mlp + 232;
    float h0[8];
#pragma unroll
    for (int h = 0; h < 8; ++h)
      h0[h] = fmaxf(W0[h * 3] * gx + W0[h * 3 + 1] * gy + W0[h * 3 + 2] * gz + B0[h], 0.0f);
    float h1[8];
#pragma unroll
    for (int h = 0; h < 8; ++h) {
      float a = B1[h];
#pragma unroll
      for (int c = 0; c < 8; ++c) a += W1[h * 8 + c] * h0[c];
      h1[h] = fmaxf(a, 0.0f);
    }
#pragma unroll
    for (int h = 0; h < 16; ++h) {
      float a = B2[h];
#pragma unroll
      for (int c = 0; c < 8; ++c) a += W2[h * 8 + c] * h1[c];
      s_wgt[k * 16 + h] = fmaxf(a, 0.0f);
    }
  }

  // --- aggregation: agg[c*16+j] = sum_k np[k][c] * wgt[k][j] ---------------
  float aggv[34];
#pragma unroll
  for (int t = 0; t < 34; ++t) {
    int   f = lane + 32 * t;
    float a = 0.0f;
    if (f < F_) {
      int c = f >> 4, j = f & 15;
#pragma unroll
      for (int kk = 0; kk < 16; ++kk) a += s_np[kk * CPAD_ + c] * s_wgt[kk * 16 + j];
    }
    aggv[t] = a;
  }
  __syncthreads();
  float* s_agg = s_union;                // [8][1072], reuses everything
#pragma unroll
  for (int t = 0; t < 34; ++t) {
    int f = lane + 32 * t;
    if (f < F_) s_agg[wid * F_ + f] = aggv[t];
  }
  __syncthreads();

  // --- phase C: WMMA GEMM  Out[16x128] = Agg[16x1072] x lin_w^T[1072x128] --
  // f32 16x16x4: A lanes 0-15 carry K{0,1}, lanes 16-31 carry K{2,3};
  // C/D: VGPR r -> row r + 8*(lane>=16), col = lane&15.
  const int col  = (wid << 4) + (lane & 15);   // wave = one 16-col N-tile
  const int hi   = lane >> 4;
  const int arow = lane & 15;                  // M row of the A fragment
  const int rrow = arow & 7;                   // clamp: rows 8..15 are zero
  const bool rowValid = (arow < 8);
  const float* lwc = lw + (size_t)col * F_;

  v8f acc = {};
  for (int kb = 0; kb < F_; kb += 4) {
    int kk = kb + 2 * hi;
    float ax = s_agg[rrow * F_ + kk];
    float ay = s_agg[rrow * F_ + kk + 1];
    v2f a; a.x = rowValid ? ax : 0.0f; a.y = rowValid ? ay : 0.0f;
    v2f bb; bb.x = lwc[kk]; bb.y = lwc[kk + 1];
    acc = __builtin_amdgcn_wmma_f32_16x16x4_f32(
        /*neg_a=*/false, a, /*neg_b=*/false, bb,
        /*c_mod=*/(short)0, acc, /*reuse_a=*/false, /*reuse_b=*/false);
  }

  const float bias = lb[col];
#pragma unroll
  for (int r = 0; r < 8; ++r) {
    int m = r + 8 * hi;
    if (m < 8) {
      int ctr = centerBase + m;
      int ob  = ctr >> 11, os = ctr & (S_ - 1);
      float v = acc[r] + bias;
      v = (v >= 0.0f) ? v : LEAKY_ * v;
      outPts[((size_t)ob * OUT_ + col) * S_ + os] = v;
    }
  }
}

// ---------------------------------------------------------------------------
extern "C" void kernel_launch(void* const* d_in, const int* in_sizes, int n_in,
                              void* d_out, int out_size, void* d_ws, size_t ws_size,
                              hipStream_t stream) {
  (void)in_sizes; (void)n_in; (void)out_size; (void)d_ws; (void)ws_size;
  const float* xyz    = (const float*)d_in[0];
  const float* points = (const float*)d_in[1];
  const float* w0     = (const float*)d_in[2];
  const float* b0     = (const float*)d_in[3];
  const float* w1     = (const float*)d_in[4];
  const float* b1     = (const float*)d_in[5];
  const float* w2     = (const float*)d_in[6];
  const float* b2     = (const float*)d_in[7];
  const float* lw     = (const float*)d_in[8];
  const float* lb     = (const float*)d_in[9];

  float* out     = (float*)d_out;
  float* newXyz  = out;                       // [B,3,S]  = 49152 floats
  float* outPts  = out + (size_t)B_ * 3 * S_; // [B,128,S]

  fps_kernel<<<B_, 1024, 0, stream>>>(xyz, newXyz);
  pointconv_kernel<<<(B_ * S_) / 8, 256, 0, stream>>>(
      xyz, points, w0, b0, w1, b1, w2, b2, lw, lb, newXyz, outPts);
}